// FeatTransKCat_54090818126592
// MI455X (gfx1250) — compile-verified
//
#include <hip/hip_runtime.h>
#include <math.h>

typedef __attribute__((ext_vector_type(2))) float v2f;
typedef __attribute__((ext_vector_type(4))) float v4f;
typedef __attribute__((ext_vector_type(8))) float v8f;

#define NBATCH 256
#define CCH    4
#define FF     256
#define NB     32
#define OC     4
#define OF     256
#define KDIM   (2 * CCH * FF)   // 2048
#define ODIM   (OC * OF)        // 1024

// ---------------- Kernel 1: S[n,c,f] = sum_nb neighbor[n,nb,c,f] ----------------
__global__ void k_nbsum(const float* __restrict__ neighbor, float* __restrict__ S) {
    int f  = threadIdx.x;                // 0..255
    int nc = blockIdx.x;                 // n*C + c
    int n = nc / CCH, c = nc % CCH;
    float acc = 0.f;
    const float* base = neighbor + ((size_t)n * NB * CCH + c) * FF + f;
#pragma unroll 8
    for (int nb = 0; nb < NB; ++nb)
        acc += __builtin_nontemporal_load(base + (size_t)nb * CCH * FF);
    S[(size_t)nc * FF + f] = acc;
}

__device__ __forceinline__ float signsqrt(float v) {
    float r = __builtin_sqrtf(fmaxf(fabsf(v), 1e-8f));
    return (v > 0.f) ? r : ((v < 0.f) ? -r : 0.f);
}
__device__ __forceinline__ float abssqrt(float v) {
    float r = __builtin_sqrtf(fmaxf(fabsf(v), 1e-8f));
    return (v == 0.f) ? 0.f : r;
}

// ---------------- Kernel 2: per (n,c): adj + ax + xcat ----------------
// t(a,b) = signsqrt(x[a]*S[b] + x[b]*S[a]);  d[b] = sum_a |t(a,b)| + 1e-7
// adj[a,b] = t(a,b)/d[b];  ax[a] = sum_b adj[a,b]*x[b]
__global__ void k_adj(const float* __restrict__ x, const float* __restrict__ S,
                      float* __restrict__ adj_out, float* __restrict__ xcat) {
    __shared__ float xr[FF];
    __shared__ float sr[FF];
    __shared__ float dinv[FF];

    int tid = threadIdx.x;               // 0..255
    int nc  = blockIdx.x;                // n*C + c
    int n = nc / CCH, c = nc % CCH;

    xr[tid] = x[(size_t)nc * FF + tid];
    sr[tid] = S[(size_t)nc * FF + tid];
    __syncthreads();

    // Pass 1: column sums (thread owns column b = tid), store reciprocal
    {
        float xb = xr[tid], sb = sr[tid];
        float d = 0.f;
#pragma unroll 4
        for (int a = 0; a < FF; ++a) {
            float v = xr[a] * sb + xb * sr[a];
            d += abssqrt(v);
        }
        dinv[tid] = 1.0f / (d + 1e-7f);
    }
    __syncthreads();

    // Pass 2: write adj with b128 non-temporal stores.
    // Thread owns 4 consecutive columns (j = tid&63 -> cols 4j..4j+3) and
    // strides rows by 4 starting at rg = tid>>6. Fully coalesced 1KB/row.
    {
        int j  = tid & 63;
        int rg = tid >> 6;
        float xb0 = xr[4*j+0], xb1 = xr[4*j+1], xb2 = xr[4*j+2], xb3 = xr[4*j+3];
        float sb0 = sr[4*j+0], sb1 = sr[4*j+1], sb2 = sr[4*j+2], sb3 = sr[4*j+3];
        float di0 = dinv[4*j+0], di1 = dinv[4*j+1], di2 = dinv[4*j+2], di3 = dinv[4*j+3];
        v4f* rowp = (v4f*)(adj_out + (size_t)nc * FF * FF) + j;
#pragma unroll 4
        for (int a = rg; a < FF; a += 4) {
            float xa = xr[a], sa = sr[a];
            v4f o;
            float t0 = signsqrt(xa * sb0 + xb0 * sa) * di0;
            float t1 = signsqrt(xa * sb1 + xb1 * sa) * di1;
            float t2 = signsqrt(xa * sb2 + xb2 * sa) * di2;
            float t3 = signsqrt(xa * sb3 + xb3 * sa) * di3;
            o[0] = (t0 != t0) ? 0.f : t0;
            o[1] = (t1 != t1) ? 0.f : t1;
            o[2] = (t2 != t2) ? 0.f : t2;
            o[3] = (t3 != t3) ? 0.f : t3;
            __builtin_nontemporal_store(o, rowp + (size_t)a * (FF / 4));
        }
    }

    // Pass 3: ax[a] (thread = row a), write xcat = [ax ; x]
    {
        float xa = xr[tid], sa = sr[tid];
        float acc = 0.f;
#pragma unroll 4
        for (int b = 0; b < FF; ++b) {
            float v  = xa * sr[b] + xr[b] * sa;
            float av = signsqrt(v) * dinv[b];
            av = (av != av) ? 0.f : av;
            acc += av * xr[b];
        }
        // xcat layout: (N, 2C, F); ax at channel c, x at channel C+c
        xcat[((size_t)n * 2 * CCH + c) * FF + tid]       = acc;
        xcat[((size_t)n * 2 * CCH + CCH + c) * FF + tid] = xa;
    }
}

// ---------------- Kernel 3: out[m,o] = sum_k xcat[m,k] * W[o,k] via WMMA f32 ----
// Block = 128 threads (4 waves); wave w owns D tile (m0..m0+15, o0+16w..o0+16w+15)
__global__ __launch_bounds__(128) void k_gemm_wmma(const float* __restrict__ xcat,
                                                   const float* __restrict__ W,
                                                   float* __restrict__ out) {
    int lane = threadIdx.x & 31;
    int wv   = threadIdx.x >> 5;
    int m0   = blockIdx.x * 16;
    int o0   = blockIdx.y * 64 + wv * 16;

    int row   = lane & 15;       // M (for A) / N (for B) index within tile
    int kh    = lane >> 4;       // lanes 16-31 carry K = 2,3
    const float2* Arow = (const float2*)(xcat + (size_t)(m0 + row) * KDIM + 2 * kh);
    const float2* Brow = (const float2*)(W    + (size_t)(o0 + row) * KDIM + 2 * kh);

    v8f acc = {};
#pragma unroll 4
    for (int k = 0; k < KDIM; k += 4) {
        float2 a2 = Arow[k >> 1];
        float2 b2 = Brow[k >> 1];
        v2f av; av[0] = a2.x; av[1] = a2.y;
        v2f bv; bv[0] = b2.x; bv[1] = b2.y;
        acc = __builtin_amdgcn_wmma_f32_16x16x4_f32(
            /*neg_a=*/false, av, /*neg_b=*/false, bv,
            /*c_mod=*/(short)0, acc, /*reuse_a=*/false, /*reuse_b=*/false);
    }

    // D layout: VGPR r, lanes 0-15 -> D[r][lane]; lanes 16-31 -> D[8+r][lane-16]
    int dn = lane & 15;
    int dm = (lane >> 4) * 8;
#pragma unroll
    for (int r = 0; r < 8; ++r)
        out[(size_t)(m0 + dm + r) * ODIM + (o0 + dn)] = acc[r];
}

extern "C" void kernel_launch(void* const* d_in, const int* in_sizes, int n_in,
                              void* d_out, int out_size, void* d_ws, size_t ws_size,
                              hipStream_t stream) {
    const float* x        = (const float*)d_in[0];   // (N,C,F)
    const float* neighbor = (const float*)d_in[1];   // (N,NB,C,F)
    const float* W        = (const float*)d_in[2];   // (OC*OF, 2C, F)

    float* out_mat = (float*)d_out;                         // (N, OC*OF)
    float* adj_out = (float*)d_out + (size_t)NBATCH * ODIM; // (N,C,F,F)

    float* S    = (float*)d_ws;                             // N*C*F floats (1 MB)
    float* xcat = S + (size_t)NBATCH * CCH * FF;            // N*2C*F floats (2 MB)

    k_nbsum<<<dim3(NBATCH * CCH), dim3(FF), 0, stream>>>(neighbor, S);
    k_adj<<<dim3(NBATCH * CCH), dim3(FF), 0, stream>>>(x, S, adj_out, xcat);
    k_gemm_wmma<<<dim3(NBATCH / 16, ODIM / 64), dim3(128), 0, stream>>>(xcat, W, out_mat);
}